// CalAttenMap_43095701848203
// MI455X (gfx1250) — compile-verified
//
#include <hip/hip_runtime.h>
#include <hip/hip_bf16.h>

#define N_NODES 2048
#define N_PAIRS 65536
#define DIM     512
#define P_OUT   16

typedef float v2f __attribute__((ext_vector_type(2)));
typedef float v8f __attribute__((ext_vector_type(8)));

// ---------------------------------------------------------------------------
// Kernel 1: node projection  out = X @ W^T + b   (X:[N,D], W:[D,D] row-major)
// One wave computes a 16(M) x 64(N) strip via V_WMMA_F32_16X16X4_F32,
// reusing the A fragment across 4 N-subtiles. K loop: 512 in steps of 4.
// ---------------------------------------------------------------------------
__global__ void proj_kernel(const float* __restrict__ X,
                            const float* __restrict__ W,
                            const float* __restrict__ bias,
                            float* __restrict__ out) {
    const int wflat = blockIdx.x * (blockDim.x >> 5) + (threadIdx.x >> 5);
    const int lane  = threadIdx.x & 31;
    const int m0 = (wflat >> 3) * 16;   // 128 M tiles
    const int n0 = (wflat & 7) * 64;    // 8 N strips of 64
    const int l15  = lane & 15;
    const int koff = (lane < 16) ? 0 : 2;

    v8f c[4] = {};
    const float* arow = X + (m0 + l15) * DIM + koff;

    for (int kb = 0; kb < DIM; kb += 4) {
        v2f a;
        a.x = arow[kb];
        a.y = arow[kb + 1];
#pragma unroll
        for (int t = 0; t < 4; ++t) {
            const float* brow = W + (n0 + t * 16 + l15) * DIM + kb + koff;
            v2f b;
            b.x = brow[0];
            b.y = brow[1];
            c[t] = __builtin_amdgcn_wmma_f32_16x16x4_f32(
                false, a, false, b, (short)0, c[t], false, false);
        }
    }

    const int orow = m0 + ((lane < 16) ? 0 : 8);
#pragma unroll
    for (int t = 0; t < 4; ++t) {
        const int col = n0 + t * 16 + l15;
        const float bv = bias[col];
#pragma unroll
        for (int r = 0; r < 8; ++r)
            out[(orow + r) * DIM + col] = c[t][r] + bv;
    }
}

// ---------------------------------------------------------------------------
// Kernel 2: per-pair logits. One wave = 16 pairs x 16 outputs (P=16 exactly
// one WMMA N tile).  A fragment built on the fly: s[i,k]*o[j,k]*union[e,k].
// B fragment: w_w^T (B[k][p] = w_w[p][k]).
// ---------------------------------------------------------------------------
__global__ void pair_kernel(const float* __restrict__ s,
                            const float* __restrict__ o,
                            const float* __restrict__ uni,
                            const int*   __restrict__ pidx,
                            const float* __restrict__ ww,
                            const float* __restrict__ wb,
                            float* __restrict__ attf) {
    const int wflat = blockIdx.x * (blockDim.x >> 5) + (threadIdx.x >> 5);
    const int lane  = threadIdx.x & 31;
    const int e0   = wflat * 16;
    const int l15  = lane & 15;
    const int koff = (lane < 16) ? 0 : 2;

    const int pr = e0 + l15;
    const int i  = pidx[pr * 2 + 0];
    const int j  = pidx[pr * 2 + 1];
    const float* sp = s + (long)i * DIM + koff;
    const float* op = o + (long)j * DIM + koff;
    const float* up = uni + (long)pr * DIM + koff;
    const float* bp = ww + (long)l15 * DIM + koff;  // column p = l15 of w_w^T

    v8f c = {};
    for (int kb = 0; kb < DIM; kb += 4) {
        v2f a, b;
        a.x = sp[kb]     * op[kb]     * up[kb];
        a.y = sp[kb + 1] * op[kb + 1] * up[kb + 1];
        b.x = bp[kb];
        b.y = bp[kb + 1];
        c = __builtin_amdgcn_wmma_f32_16x16x4_f32(
            false, a, false, b, (short)0, c, false, false);
    }

    const float bv = wb[l15];
    const int prow = e0 + ((lane < 16) ? 0 : 8);
#pragma unroll
    for (int r = 0; r < 8; ++r)
        attf[(prow + r) * P_OUT + l15] = c[r] + bv;
}

// ---------------------------------------------------------------------------
// Binning helpers: group pair indices by source node (idx0).
// ---------------------------------------------------------------------------
__global__ void zero_int_kernel(int* p, int n) {
    int t = blockIdx.x * blockDim.x + threadIdx.x;
    if (t < n) p[t] = 0;
}

__global__ void count_kernel(const int* __restrict__ pidx, int* __restrict__ counts) {
    int e = blockIdx.x * blockDim.x + threadIdx.x;
    if (e < N_PAIRS) atomicAdd(&counts[pidx[e * 2]], 1);
}

__global__ void scan_kernel(const int* __restrict__ counts, int* __restrict__ offsets) {
    if (blockIdx.x == 0 && threadIdx.x == 0) {
        int acc = 0;
        for (int i = 0; i < N_NODES; ++i) { offsets[i] = acc; acc += counts[i]; }
        offsets[N_NODES] = acc;
    }
}

__global__ void fill_kernel(const int* __restrict__ pidx,
                            const int* __restrict__ offsets,
                            int* __restrict__ cursor,
                            int* __restrict__ bins) {
    int e = blockIdx.x * blockDim.x + threadIdx.x;
    if (e < N_PAIRS) {
        int i = pidx[e * 2];
        int slot = atomicAdd(&cursor[i], 1);
        bins[offsets[i] + slot] = e;
    }
}

// ---------------------------------------------------------------------------
// Kernel 3: per-row fused scatter + diag-mask + softmax.
// One block per source node i.  The full 2048x16 row slice (128 KB) lives in
// LDS (320 KB per WGP), so the dense pre-softmax tensor never touches HBM:
// d_out is written exactly once.
// ---------------------------------------------------------------------------
__global__ void softmax_kernel(const float* __restrict__ attf,
                               const int*   __restrict__ pidx,
                               const int*   __restrict__ offsets,
                               const int*   __restrict__ bins,
                               float* __restrict__ out) {
    extern __shared__ float row[];          // N_NODES * P_OUT floats = 128 KB
    __shared__ float red[32 * P_OUT];
    __shared__ float maxv[P_OUT], sumv[P_OUT];

    const int i = blockIdx.x;
    const int t = threadIdx.x;              // 512 threads

    // zero the row slice
    for (int idx = t; idx < N_NODES * P_OUT; idx += 512) row[idx] = 0.0f;
    __syncthreads();

    // scatter binned pair logits into LDS (LDS atomics; pairs may repeat (i,j))
    const int beg = offsets[i];
    const int cnt = offsets[i + 1] - beg;
    for (int w = t; w < cnt * P_OUT; w += 512) {
        const int e = bins[beg + (w >> 4)];
        const int p = w & 15;
        const int j = pidx[e * 2 + 1];
        atomicAdd(&row[j * P_OUT + p], attf[e * P_OUT + p]);
    }
    __syncthreads();

    // diagonal mask
    if (t < P_OUT) row[i * P_OUT + t] -= 10000.0f;
    __syncthreads();

    const int p = t & 15;
    const int g = t >> 4;                   // 32 groups per channel

    // per-channel max over j
    float m = -3.0e38f;
    for (int j = g; j < N_NODES; j += 32) m = fmaxf(m, row[j * P_OUT + p]);
    red[g * P_OUT + p] = m;
    __syncthreads();
    if (t < P_OUT) {
        float mm = red[t];
        for (int gg = 1; gg < 32; ++gg) mm = fmaxf(mm, red[gg * P_OUT + t]);
        maxv[t] = mm;
    }
    __syncthreads();

    // exp + per-channel sum (exp written back into LDS)
    const float mv = maxv[p];
    float sm = 0.0f;
    for (int j = g; j < N_NODES; j += 32) {
        float v = __expf(row[j * P_OUT + p] - mv);
        row[j * P_OUT + p] = v;
        sm += v;
    }
    red[g * P_OUT + p] = sm;
    __syncthreads();
    if (t < P_OUT) {
        float ss = 0.0f;
        for (int gg = 0; gg < 32; ++gg) ss += red[gg * P_OUT + t];
        sumv[t] = ss;
    }
    __syncthreads();

    // normalize and write out once, coalesced. (t + k*512) & 15 == t & 15.
    const float inv = 1.0f / sumv[t & 15];
    const size_t base = (size_t)i * (N_NODES * P_OUT);
    for (int idx = t; idx < N_NODES * P_OUT; idx += 512)
        out[base + idx] = row[idx] * inv;
}

// ---------------------------------------------------------------------------
extern "C" void kernel_launch(void* const* d_in, const int* in_sizes, int n_in,
                              void* d_out, int out_size, void* d_ws, size_t ws_size,
                              hipStream_t stream) {
    const float* obj  = (const float*)d_in[0];
    const float* uni  = (const float*)d_in[1];
    const int*   pidx = (const int*)  d_in[2];
    const float* ws_w = (const float*)d_in[3];
    const float* ws_b = (const float*)d_in[4];
    const float* wo_w = (const float*)d_in[5];
    const float* wo_b = (const float*)d_in[6];
    const float* w_w  = (const float*)d_in[7];
    const float* w_b  = (const float*)d_in[8];
    float* outp = (float*)d_out;

    // workspace carve-up (~12.3 MB)
    float* s    = (float*)d_ws;
    float* o    = s + (size_t)N_NODES * DIM;
    float* attf = o + (size_t)N_NODES * DIM;
    int* counts  = (int*)(attf + (size_t)N_PAIRS * P_OUT);
    int* offsets = counts + N_NODES;
    int* cursor  = offsets + (N_NODES + 1);
    int* bins    = cursor + N_NODES;

    // 1) node projections (WMMA f32): 1024 waves each = 128 blocks x 8 waves
    proj_kernel<<<128, 256, 0, stream>>>(obj, ws_w, ws_b, s);
    proj_kernel<<<128, 256, 0, stream>>>(obj, wo_w, wo_b, o);

    // 2) pair logits (WMMA f32): 4096 waves = 512 blocks x 8 waves
    pair_kernel<<<512, 256, 0, stream>>>(s, o, uni, pidx, w_w, w_b, attf);

    // 3) bin pairs by source node
    zero_int_kernel<<<(N_NODES + 255) / 256, 256, 0, stream>>>(counts, N_NODES);
    zero_int_kernel<<<(N_NODES + 255) / 256, 256, 0, stream>>>(cursor, N_NODES);
    count_kernel<<<N_PAIRS / 256, 256, 0, stream>>>(pidx, counts);
    scan_kernel<<<1, 1, 0, stream>>>(counts, offsets);
    fill_kernel<<<N_PAIRS / 256, 256, 0, stream>>>(pidx, offsets, cursor, bins);

    // 4) fused scatter + mask + softmax; 128 KB dynamic LDS row slice/block
    softmax_kernel<<<N_NODES, 512, N_NODES * P_OUT * sizeof(float), stream>>>(
        attf, pidx, offsets, bins, outp);
}